// MAHGAEncoder_55920474194408
// MI455X (gfx1250) — compile-verified
//
#include <hip/hip_runtime.h>
#include <math.h>
#include <stdint.h>

// MAHGA encoder for MI455X (gfx1250, wave32).
// Dense x@W GEMMs + final FC use V_WMMA_F32_16X16X4_F32 (full fp32 precision;
// the workload is L2-gather/atomic bound, so fp32 matrix ops are free).
// A-tiles are staged into LDS once per block via the gfx1250 async
// Global->LDS path (GLOBAL_LOAD_ASYNC_TO_LDS_B128 + s_wait_asynccnt),
// removing the 8x redundant per-wave A traffic.
// Edge aggregation uses float4 gathers from L2-resident xw and f32 atomics.

#define NN 50000
#define NE 800000
#define HC 128
#define OC 64

typedef __attribute__((ext_vector_type(2))) float v2f;
typedef __attribute__((ext_vector_type(8))) float v8f;

// Async 16-byte Global->LDS copy (gfx1250). INST_OFFSET is added to BOTH the
// LDS and global addresses, so one address pair serves consecutive chunks.
// Generic pointers to __shared__ have addr[31:0] == LDS byte offset (ISA
// aperture rule), so the truncated pointer is a valid LDS address VGPR.
__device__ __forceinline__ void async_g2l_b128(uint32_t lds_off, const float* g) {
  asm volatile("global_load_async_to_lds_b128 %0, %1, off"
               :: "v"(lds_off), "v"(g) : "memory");
}
__device__ __forceinline__ void async_g2l_b128_o16(uint32_t lds_off, const float* g) {
  asm volatile("global_load_async_to_lds_b128 %0, %1, off offset:16"
               :: "v"(lds_off), "v"(g) : "memory");
}
__device__ __forceinline__ void wait_async0() {
  asm volatile("s_wait_asynccnt 0x0" ::: "memory");
}

// ---------------- degree / norm ----------------
__global__ __launch_bounds__(256) void k_fill_deg(float* __restrict__ deg) {
  int i = blockIdx.x * 256 + threadIdx.x;
  if (i < NN) deg[i] = 1.0f;  // self loop
}

__global__ __launch_bounds__(256) void k_count_deg(const long long* __restrict__ dst,
                                                   float* __restrict__ deg) {
  int e = blockIdx.x * 256 + threadIdx.x;
  if (e < NE) atomicAdd(&deg[dst[e]], 1.0f);
}

__global__ __launch_bounds__(256) void k_dinv(float* __restrict__ deg) {
  int i = blockIdx.x * 256 + threadIdx.x;
  if (i < NN) deg[i] = rsqrtf(deg[i]);  // deg >= 1 always
}

// ---------------- dense GEMM: [NN x 128] @ [128 x 128] via f32 WMMA ----------------
// Block = 256 threads = 8 waves; block handles one 16-row M-tile, wave w handles
// N-tile w (8 tiles * 16 = 128 cols). The 16x128 A tile (8 KB) is staged into
// LDS once per block with async B128 copies, then all 8 waves feed their WMMAs
// from LDS (ds_load_2addr_b64). K looped in steps of 4 -> 32 WMMAs/tile.
// A frag (16x4 f32): lanes 0-15 row M=lane, K={k0,k0+1}; lanes 16-31 same rows,
// K={k0+2,k0+3}. B frag (4x16 f32): mirrored lane layout over N.
#define ASTRIDE 132  // row stride in floats: bank = (4m + k) % 64 -> conflict free
template <bool RELU_IN>
__global__ __launch_bounds__(256) void k_gemm128(const float* __restrict__ A,
                                                 const float* __restrict__ W,
                                                 float* __restrict__ out) {
  __shared__ float As[16 * ASTRIDE];
  const int tid = threadIdx.x;

  // cooperative async stage: thread tid copies 8 floats (two B128 chunks)
  {
    const int r = tid >> 4;          // 0..15 rows
    const int c = (tid & 15) * 8;    // 0..120 cols, step 8
    const float* g = A + (size_t)(blockIdx.x * 16 + r) * 128 + c;
    const uint32_t l = (uint32_t)(uintptr_t)(&As[r * ASTRIDE + c]);
    async_g2l_b128(l, g);        // floats c..c+3
    async_g2l_b128_o16(l, g);    // floats c+4..c+7
    wait_async0();               // drain this wave's ASYNCcnt
  }
  __syncthreads();               // publish LDS tile to all 8 waves

  const int lane = tid & 31;
  const int wave = tid >> 5;
  const int m16  = lane & 15;
  const int kh   = lane >> 4;    // 0 or 1 -> k offset 2*kh
  const int colB = wave * 16 + m16;
  v8f acc = {};
#pragma unroll
  for (int kk = 0; kk < 128; kk += 4) {
    const int k0 = kk + 2 * kh;
    float a0 = As[m16 * ASTRIDE + k0];
    float a1 = As[m16 * ASTRIDE + k0 + 1];
    if (RELU_IN) { a0 = fmaxf(a0, 0.0f); a1 = fmaxf(a1, 0.0f); }
    v2f a, b;
    a.x = a0; a.y = a1;
    b.x = W[k0 * 128 + colB];
    b.y = W[(k0 + 1) * 128 + colB];
    acc = __builtin_amdgcn_wmma_f32_16x16x4_f32(false, a, false, b, (short)0, acc,
                                                false, false);
  }
#pragma unroll
  for (int i = 0; i < 8; i++) {
    const int m = (lane < 16) ? i : (i + 8);
    out[(blockIdx.x * 16 + m) * 128 + wave * 16 + m16] = acc[i];
  }
}

// ---------------- out = bias + self-loop term ----------------
__global__ __launch_bounds__(256) void k_agg_init(const float* __restrict__ xw,
                                                  const float* __restrict__ dinv,
                                                  const float* __restrict__ bias,
                                                  float* __restrict__ out) {
  const int idx = blockIdx.x * 256 + threadIdx.x;  // 6.4M exact
  const int i = idx >> 7, c = idx & 127;
  const float di = dinv[i];
  out[idx] = bias[c] + xw[idx] * di * di;
}

// ---------------- edge scatter: one wave per edge, float4 per lane ----------------
__global__ __launch_bounds__(256) void k_scatter(const float* __restrict__ xw,
                                                 const float* __restrict__ dinv,
                                                 const long long* __restrict__ ei,
                                                 float* __restrict__ out) {
  const int e = blockIdx.x * 8 + (threadIdx.x >> 5);  // 100000*8 = NE exact
  const int lane = threadIdx.x & 31;
  const long long s = ei[e];
  const long long d = ei[NE + e];
  const float w = dinv[s] * dinv[d];
  const float4 v = *(const float4*)(xw + (size_t)s * 128 + lane * 4);
  float* o = out + (size_t)d * 128 + lane * 4;
  atomicAdd(o + 0, v.x * w);
  atomicAdd(o + 1, v.y * w);
  atomicAdd(o + 2, v.z * w);
  atomicAdd(o + 3, v.w * w);
}

// ---------------- fused attention + concat + FC (LDS-fed WMMA) ----------------
// Block = 256 threads handles 16 nodes. Phase 1: 16-thread groups compute
// ReLU(h_struct), ReLU(h1/h2), 2-way attention softmax, and write the
// [16 x 256] combined matrix to LDS (row stride 260 -> conflict free).
// Phase 2: waves 0..3 each compute one 16x16 output tile over K=256 (64 WMMAs).
__global__ __launch_bounds__(256) void k_fuse(const float* __restrict__ hs,
                                              const float* __restrict__ h1,
                                              const float* __restrict__ h2,
                                              const float* __restrict__ Watt,
                                              const float* __restrict__ batt,
                                              const float* __restrict__ Wfc,
                                              const float* __restrict__ bfc,
                                              float* __restrict__ out) {
  __shared__ float comb[16 * 260];
  const int tid = threadIdx.x;
  const int g = tid >> 4;   // node in tile (0..15)
  const int t = tid & 15;   // channel group (0..15), 8 channels each
  const int node = blockIdx.x * 16 + g;
  const size_t base = (size_t)node * 128;

  float a1[8], a2[8], s0[8];
  float p1 = 0.0f, p2 = 0.0f;
#pragma unroll
  for (int j = 0; j < 8; j++) {
    const int c = t * 8 + j;
    const float x1 = fmaxf(h1[base + c], 0.0f);
    const float x2 = fmaxf(h2[base + c], 0.0f);
    s0[j] = fmaxf(hs[base + c], 0.0f);
    a1[j] = x1; a2[j] = x2;
    const float wa = Watt[c];
    p1 += x1 * wa;
    p2 += x2 * wa;
  }
#pragma unroll
  for (int off = 8; off > 0; off >>= 1) {
    p1 += __shfl_xor(p1, off, 16);
    p2 += __shfl_xor(p2, off, 16);
  }
  const float sc1 = p1 + batt[0];
  const float sc2 = p2 + batt[0];
  const float mx = fmaxf(sc1, sc2);
  const float e1 = __expf(sc1 - mx);
  const float e2 = __expf(sc2 - mx);
  const float inv = 1.0f / (e1 + e2);
  const float w1 = e1 * inv, w2 = e2 * inv;
#pragma unroll
  for (int j = 0; j < 8; j++) {
    const int c = t * 8 + j;
    comb[g * 260 + c] = s0[j];                          // h_struct half
    comb[g * 260 + 128 + c] = w1 * a1[j] + w2 * a2[j];  // h_meta half
  }
  __syncthreads();

  const int wave = tid >> 5;
  if (wave < 4) {  // uniform per wave -> EXEC all ones inside
    const int lane = tid & 31;
    const int m16 = lane & 15;
    const int kh = lane >> 4;
    v8f acc = {};
#pragma unroll 4
    for (int kk = 0; kk < 256; kk += 4) {
      const int k0 = kk + 2 * kh;
      v2f a, b;
      a.x = comb[m16 * 260 + k0];
      a.y = comb[m16 * 260 + k0 + 1];
      b.x = Wfc[k0 * 64 + wave * 16 + m16];
      b.y = Wfc[(k0 + 1) * 64 + wave * 16 + m16];
      acc = __builtin_amdgcn_wmma_f32_16x16x4_f32(false, a, false, b, (short)0, acc,
                                                  false, false);
    }
#pragma unroll
    for (int i = 0; i < 8; i++) {
      const int m = (lane < 16) ? i : (i + 8);
      const int n = wave * 16 + m16;
      out[(size_t)(blockIdx.x * 16 + m) * 64 + n] = acc[i] + bfc[n];
    }
  }
}

extern "C" void kernel_launch(void* const* d_in, const int* in_sizes, int n_in,
                              void* d_out, int out_size, void* d_ws, size_t ws_size,
                              hipStream_t stream) {
  (void)in_sizes; (void)n_in; (void)out_size; (void)ws_size;
  const float* x        = (const float*)d_in[0];
  const long long* ei   = (const long long*)d_in[1];  // int64 [2, NE]
  const float* W_s1     = (const float*)d_in[2];
  const float* b_s1     = (const float*)d_in[3];
  const float* W_s2     = (const float*)d_in[4];
  const float* b_s2     = (const float*)d_in[5];
  const float* W_m1     = (const float*)d_in[6];
  const float* b_m1     = (const float*)d_in[7];
  const float* W_m21    = (const float*)d_in[8];
  const float* b_m21    = (const float*)d_in[9];
  const float* W_m22    = (const float*)d_in[10];
  const float* b_m22    = (const float*)d_in[11];
  const float* W_att    = (const float*)d_in[12];
  const float* b_att    = (const float*)d_in[13];
  const float* W_fc     = (const float*)d_in[14];
  const float* b_fc     = (const float*)d_in[15];
  float* out = (float*)d_out;

  char* ws = (char*)d_ws;
  const size_t BUF = (size_t)NN * HC * sizeof(float);  // 25.6 MB
  float* dinv = (float*)ws;                            // 200 KB slot
  float* bufX = (float*)(ws + 256 * 1024);             // xw scratch
  float* bufB = (float*)(ws + 256 * 1024 + BUF);       // t / h1
  float* bufC = (float*)(ws + 256 * 1024 + 2 * BUF);   // h_struct (pre-relu)
  float* bufD = (float*)(ws + 256 * 1024 + 3 * BUF);   // t2 / h2 (pre-relu)

  const dim3 B256(256);

  // symmetric normalization coefficients
  k_fill_deg<<<(NN + 255) / 256, B256, 0, stream>>>(dinv);
  k_count_deg<<<NE / 256, B256, 0, stream>>>(ei + NE, dinv);
  k_dinv<<<(NN + 255) / 256, B256, 0, stream>>>(dinv);

  auto gcn = [&](const float* in, const float* W, const float* b, float* o,
                 bool relu_in) {
    if (relu_in)
      k_gemm128<true><<<NN / 16, B256, 0, stream>>>(in, W, bufX);
    else
      k_gemm128<false><<<NN / 16, B256, 0, stream>>>(in, W, bufX);
    k_agg_init<<<(NN * HC) / 256, B256, 0, stream>>>(bufX, dinv, b, o);
    k_scatter<<<NE / 8, B256, 0, stream>>>(bufX, dinv, ei, o);
  };

  // structural branch (relu fused into consumer loads)
  gcn(x, W_s1, b_s1, bufB, false);    // t = gcn1(x)
  gcn(bufB, W_s2, b_s2, bufC, true);  // h_struct = gcn2(relu(t))
  // meta branch 1
  gcn(x, W_m1, b_m1, bufB, false);    // h1
  // meta branch 2
  gcn(x, W_m21, b_m21, bufD, false);  // t2
  gcn(bufD, W_m22, b_m22, bufD, true);// h2 = gcn(relu(t2))

  // attention softmax fusion + concat + FC
  k_fuse<<<NN / 16, B256, 0, stream>>>(bufC, bufB, bufD, W_att, b_att, W_fc, b_fc,
                                       out);
}